// MSTParserLSTMModel_43980465111474
// MI455X (gfx1250) — compile-verified
//
#include <hip/hip_runtime.h>
#include <hip/hip_bf16.h>

// ---- problem constants (match reference) ----
#define TT 512   // tokens
#define DD 256   // embed dim
#define LL 256   // lstm dim
#define HH 512   // hidden units
#define GG 1024  // 4*L gate width
#define C2 512   // 2*L concat width

typedef float v2f __attribute__((ext_vector_type(2)));
typedef float v8f __attribute__((ext_vector_type(8)));

// ---- fast activations (gfx1250 v_tanh_f32 when available) ----
__device__ __forceinline__ float fast_tanh(float x) {
#if __has_builtin(__builtin_amdgcn_tanhf)
  return __builtin_amdgcn_tanhf(x);
#else
  return tanhf(x);
#endif
}
__device__ __forceinline__ float fast_sigmoid(float x) {
  // exact identity: sigma(x) = 0.5*tanh(x/2) + 0.5
  return __builtin_fmaf(0.5f, fast_tanh(0.5f * x), 0.5f);
}

// ---- CDNA5 async global->LDS copy (ASYNCcnt-tracked, no VGPR bounce) ----
// LDS destination is the low 32 bits of the flat pointer (aperture rules:
// flat LDS address = SHARED_BASE[63:32] | lds_offset).
__device__ __forceinline__ void async_load_b128(const float* gsrc, float* lds_dst) {
  const unsigned lds_off = (unsigned)(size_t)lds_dst;
  asm volatile("global_load_async_to_lds_b128 %0, %1, off"
               :: "v"(lds_off), "v"(gsrc)
               : "memory");
}
__device__ __forceinline__ void wait_async0() {
  asm volatile("s_wait_asynccnt 0" ::: "memory");
}

// =====================================================================
// GEMM: C[M,N] = A[M,K] @ B[K,N] (+ bias[N] if bias != null), fp32 WMMA.
// Block = 128 threads = 4 waves; each wave computes one 16x16 C tile;
// block covers 16 rows x 64 cols. Requires M%16==0, N%64==0, K%4==0.
// Fragment layouts per CDNA5 ISA 7.12.2 (wave32):
//   A 16x4 f32 : lane m=l&15, half=l>>4 -> VGPR0/1 = A[m][k+2*half +0/1]
//   B 4x16 f32 : lane n=l&15, half=l>>4 -> VGPR0/1 = B[k+2*half +0/1][n]
//   C/D 16x16  : VGPR r, lane l -> C[r + 8*(l>>4)][l&15]
// =====================================================================
__global__ void __launch_bounds__(128)
gemm_wmma_f32(const float* __restrict__ A, const float* __restrict__ B,
              const float* __restrict__ bias, float* __restrict__ C,
              int M, int K, int N)
{
  const int lane  = threadIdx.x & 31;
  const int wave  = threadIdx.x >> 5;
  const int row0  = blockIdx.y * 16;
  const int col0  = blockIdx.x * 64 + wave * 16;
  const int mn    = lane & 15;     // M index for A frag, N index for B frag
  const int khalf = lane >> 4;     // 0 or 1

  if (row0 >= M || col0 >= N) return;   // wave-uniform: EXEC stays all-1s

  v8f acc = {};
  const float* __restrict__ arow = A + (size_t)(row0 + mn) * K;
  for (int k = 0; k < K; k += 4) {
    const int ka = k + 2 * khalf;
    v2f a = *(const v2f*)(arow + ka);                 // 8B aligned: ka even
    v2f b;
    b.x = B[(size_t)ka * N + col0 + mn];
    b.y = B[(size_t)(ka + 1) * N + col0 + mn];
    acc = __builtin_amdgcn_wmma_f32_16x16x4_f32(
        /*neg_a=*/false, a, /*neg_b=*/false, b,
        /*c_mod=*/(short)0, acc, /*reuse_a=*/false, /*reuse_b=*/false);
  }

  const int n     = lane & 15;
  const int rbase = (lane >> 4) * 8;
  const float bv  = bias ? bias[col0 + n] : 0.0f;
#pragma unroll
  for (int r = 0; r < 8; ++r) {
    C[(size_t)(row0 + rbase + r) * N + col0 + n] = acc[r] + bv;
  }
}

// =====================================================================
// Bidirectional LSTM layer scan. grid.x = 2 (block 0 = forward,
// block 1 = backward over reversed sequence). 1024 threads: thread n owns
// gate column n of z = XW[t] + h@U  (bias already folded into XW).
// Threads < LL compose gates (order i,f,g,o) and write h into the concat
// output V[t][colOff + j]; backward writes at row T-1-t, which realizes
// the reference's rb[::-1] pairing directly.
// =====================================================================
__global__ void __launch_bounds__(1024)
lstm_bidir(const float* __restrict__ XWf, const float* __restrict__ Uf,
           const float* __restrict__ XWb, const float* __restrict__ Ub,
           float* __restrict__ V /* [T, C2] */)
{
  const int rev = blockIdx.x;
  const float* __restrict__ XW = rev ? XWb : XWf;
  const float* __restrict__ U  = rev ? Ub  : Uf;
  const int colOff = rev ? LL : 0;

  __shared__ float hs[LL];
  __shared__ float zs[GG];

  const int n = threadIdx.x;   // 0..1023
  if (n < LL) hs[n] = 0.0f;
  float c = 0.0f;              // cell state, valid for n < LL
  __syncthreads();

  for (int s = 0; s < TT; ++s) {
    const int t = rev ? (TT - 1 - s) : s;
    // z_n = XW[t][n] + sum_k hs[k] * U[k][n]
    float z = XW[(size_t)t * GG + n];
    if (s + 1 < TT) {
      const int tn = rev ? (t - 1) : (t + 1);
      __builtin_prefetch(XW + (size_t)tn * GG + n, 0, 1);
    }
#pragma unroll 8
    for (int k = 0; k < LL; ++k) {
      z = __builtin_fmaf(hs[k], U[(size_t)k * GG + n], z);
    }
    zs[n] = z;
    __syncthreads();
    if (n < LL) {
      const float ig = fast_sigmoid(zs[n]);
      const float fg = fast_sigmoid(zs[LL + n]);
      const float gg = fast_tanh  (zs[2 * LL + n]);
      const float og = fast_sigmoid(zs[3 * LL + n]);
      c = __builtin_fmaf(fg, c, ig * gg);
      const float h = og * fast_tanh(c);
      hs[n] = h;
      V[(size_t)t * C2 + colOff + n] = h;
    }
    __syncthreads();
  }
}

// =====================================================================
// Pairwise head: scores[i,j] = sum_h w[h]*tanh(HF[i,h] + MF[j,h]) + outBias
// (hidBias already folded into HF). Block = 256 threads computes a 16x16
// score tile. H is chunked by 128 through double-buffered LDS filled with
// GLOBAL_LOAD_ASYNC_TO_LDS_B128 (ASYNCcnt): chunk c+1 streams in while
// chunk c's 128 tanh+fma per thread execute — copy latency fully hidden.
// Padded pitch (132 words) keeps column reads bank-conflict-free.
// =====================================================================
#define HCHUNK 128
#define LPITCH 132
#define NCHUNK (HH / HCHUNK)

__device__ __forceinline__ void issue_chunk(const float* __restrict__ HF,
                                            const float* __restrict__ MF,
                                            int i0, int j0, int hc,
                                            float* hbuf, float* mbuf, int tid)
{
  // 16 rows x 128 floats = 512 float4 per matrix; 2 async b128 per thread each
  for (int v = tid; v < 16 * (HCHUNK / 4); v += 256) {
    const int r  = v >> 5;        // / 32
    const int c4 = (v & 31) * 4;
    async_load_b128(HF + (size_t)(i0 + r) * HH + hc + c4, &hbuf[r * LPITCH + c4]);
    async_load_b128(MF + (size_t)(j0 + r) * HH + hc + c4, &mbuf[r * LPITCH + c4]);
  }
}

__global__ void __launch_bounds__(256)
pairwise_head(const float* __restrict__ HF, const float* __restrict__ MF,
              const float* __restrict__ w, const float* __restrict__ outBias,
              float* __restrict__ scores)
{
  __shared__ __align__(16) float hfs[2][16 * LPITCH];
  __shared__ __align__(16) float mfs[2][16 * LPITCH];
  __shared__ float ws[HH];

  const int i0  = blockIdx.y * 16;
  const int j0  = blockIdx.x * 16;
  const int tid = threadIdx.x;
  const int ti  = tid >> 4;
  const int tj  = tid & 15;

  // preload the output weight vector and kick off chunk 0
  for (int v = tid; v < HH; v += 256) ws[v] = w[v];
  issue_chunk(HF, MF, i0, j0, 0, hfs[0], mfs[0], tid);

  float acc = 0.0f;
  for (int ci = 0; ci < NCHUNK; ++ci) {
    wait_async0();       // this wave's async fills for chunk ci complete
    __syncthreads();     // all waves done filling ci AND done reading ci-1
    if (ci + 1 < NCHUNK) {
      issue_chunk(HF, MF, i0, j0, (ci + 1) * HCHUNK,
                  hfs[(ci + 1) & 1], mfs[(ci + 1) & 1], tid);
    }
    const float* __restrict__ hb = hfs[ci & 1];
    const float* __restrict__ mb = mfs[ci & 1];
    const float* __restrict__ wc = ws + ci * HCHUNK;
#pragma unroll 8
    for (int h = 0; h < HCHUNK; ++h) {
      const float t = fast_tanh(hb[ti * LPITCH + h] + mb[tj * LPITCH + h]);
      acc = __builtin_fmaf(wc[h], t, acc);
    }
    __syncthreads();     // everyone done with buffer (ci&1) before refill at ci+2
  }
  scores[(size_t)(i0 + ti) * TT + (j0 + tj)] = acc + outBias[0];
}

// =====================================================================
// Launch
// =====================================================================
extern "C" void kernel_launch(void* const* d_in, const int* in_sizes, int n_in,
                              void* d_out, int out_size, void* d_ws, size_t ws_size,
                              hipStream_t stream) {
  (void)in_sizes; (void)n_in; (void)out_size; (void)ws_size;

  const float* X     = (const float*)d_in[0];   // [T, D] (leading 1 squeezed)
  const float* W_f1  = (const float*)d_in[1];
  const float* U_f1  = (const float*)d_in[2];
  const float* b_f1  = (const float*)d_in[3];
  const float* W_b1  = (const float*)d_in[4];
  const float* U_b1  = (const float*)d_in[5];
  const float* b_b1  = (const float*)d_in[6];
  const float* W_f2  = (const float*)d_in[7];
  const float* U_f2  = (const float*)d_in[8];
  const float* b_f2  = (const float*)d_in[9];
  const float* W_b2  = (const float*)d_in[10];
  const float* U_b2  = (const float*)d_in[11];
  const float* b_b2  = (const float*)d_in[12];
  const float* FOH   = (const float*)d_in[13];
  const float* FOM   = (const float*)d_in[14];
  const float* hidB  = (const float*)d_in[15];
  const float* outW  = (const float*)d_in[16];
  const float* outB  = (const float*)d_in[17];
  float* scores = (float*)d_out;

  // workspace layout (floats): XWa/XWb reused between layers
  float* ws  = (float*)d_ws;
  float* XWa = ws;                        // T*GG
  float* XWb = XWa + (size_t)TT * GG;     // T*GG
  float* V   = XWb + (size_t)TT * GG;     // T*C2
  float* Hc  = V   + (size_t)TT * C2;     // T*C2
  float* HF  = Hc  + (size_t)TT * C2;     // T*HH
  float* MF  = HF  + (size_t)TT * HH;     // T*HH

  const dim3 gemmBlk(128);
  const dim3 gGate(GG / 64, TT / 16);     // N=1024 tiles
  const dim3 gHead(HH / 64, TT / 16);     // N=512 tiles

  // --- layer 1: input projections, then bidir scan into V ---
  gemm_wmma_f32<<<gGate, gemmBlk, 0, stream>>>(X, W_f1, b_f1, XWa, TT, DD, GG);
  gemm_wmma_f32<<<gGate, gemmBlk, 0, stream>>>(X, W_b1, b_b1, XWb, TT, DD, GG);
  lstm_bidir<<<dim3(2), dim3(1024), 0, stream>>>(XWa, U_f1, XWb, U_b1, V);

  // --- layer 2: projections from V, bidir scan into Hc ---
  gemm_wmma_f32<<<gGate, gemmBlk, 0, stream>>>(V, W_f2, b_f2, XWa, TT, C2, GG);
  gemm_wmma_f32<<<gGate, gemmBlk, 0, stream>>>(V, W_b2, b_b2, XWb, TT, C2, GG);
  lstm_bidir<<<dim3(2), dim3(1024), 0, stream>>>(XWa, U_f2, XWb, U_b2, Hc);

  // --- head projections (hidBias folded into HF) ---
  gemm_wmma_f32<<<gHead, gemmBlk, 0, stream>>>(Hc, FOH, hidB,    HF, TT, C2, HH);
  gemm_wmma_f32<<<gHead, gemmBlk, 0, stream>>>(Hc, FOM, nullptr, MF, TT, C2, HH);

  // --- pairwise tanh-weighted scores ---
  pairwise_head<<<dim3(TT / 16, TT / 16), dim3(256), 0, stream>>>(
      HF, MF, outW, outB, scores);
}